// ReprogrammingFuntion_38706245271901
// MI455X (gfx1250) — compile-verified
//
#include <hip/hip_runtime.h>
#include <hip/hip_bf16.h>
#include <math.h>

typedef __attribute__((ext_vector_type(4))) float v4f;
typedef __attribute__((ext_vector_type(2))) float v2f;
typedef __attribute__((ext_vector_type(8))) float v8f;

#define SEQ_LEN     128
#define EMB_D       768
#define OUT_ELEMS   (128 * 3 * 384 * 384)   /* 56,623,104 */
#define VEC_TOTAL   (OUT_ELEMS / 4)         /* 14,155,776 float4 */
#define BLOCK1      256
#define GRID1       (VEC_TOTAL / BLOCK1)    /* 55,296 blocks -> 55,296 partials (divisible by 32) */

// Kernel 1: fused gather + tanh + patch-replicate-transpose + clip, with
// per-block sum-of-squares partials. One float4 (global_load_b128 /
// nontemporal global_store_b128) per thread, fully coalesced.
__global__ __launch_bounds__(BLOCK1)
void emb_img_kernel(const int* __restrict__ tok,
                    const float* __restrict__ emb,
                    float* __restrict__ out,
                    float* __restrict__ partial) {
    const unsigned g  = blockIdx.x * (unsigned)BLOCK1 + threadIdx.x;  // float4 index
    const unsigned x4 = g % 96u;            // 384/4 float4 per row
    const unsigned t  = g / 96u;
    const unsigned y  = t % 384u;
    const unsigned t2 = t / 384u;
    const unsigned c  = t2 % 3u;
    const unsigned n  = t2 / 3u;

    const unsigned px  = x4 >> 2;           // patch column 0..23
    const unsigned ix0 = (x4 & 3u) << 2;    // 0,4,8,12 inside patch
    const unsigned py  = y >> 4;            // patch row 0..23
    const unsigned iy  = y & 15u;

    const unsigned p   = py * 24u + px;
    const unsigned src = (p < (SEQ_LEN - 1u)) ? p : (SEQ_LEN - 1u);
    const unsigned token = (unsigned)tok[n * SEQ_LEN + src];

    const size_t eoff = (size_t)token * EMB_D + c * 256u + iy * 16u + ix0;
    const v4f ev = *(const v4f*)(emb + eoff);   // 16B-aligned gather

    const float a0 = tanhf(ev.x);
    const float a1 = tanhf(ev.y);
    const float a2 = tanhf(ev.z);
    const float a3 = tanhf(ev.w);

    // pert_norm accumulates the PRE-clip image, per reference
    float s = a0 * a0 + a1 * a1 + a2 * a2 + a3 * a3;

    // out = (clip(img*0.5+0.5, 0, 1) - 0.5) / 0.5  (mirror reference fp ops)
    v4f r;
    {
        float u;
        u = fminf(fmaxf(a0 * 0.5f + 0.5f, 0.0f), 1.0f); r.x = (u - 0.5f) * 2.0f;
        u = fminf(fmaxf(a1 * 0.5f + 0.5f, 0.0f), 1.0f); r.y = (u - 0.5f) * 2.0f;
        u = fminf(fmaxf(a2 * 0.5f + 0.5f, 0.0f), 1.0f); r.z = (u - 0.5f) * 2.0f;
        u = fminf(fmaxf(a3 * 0.5f + 0.5f, 0.0f), 1.0f); r.w = (u - 0.5f) * 2.0f;
    }
    // NT store: 226 MB write stream must not evict the 154 MB gather table from L2
    __builtin_nontemporal_store(r, (v4f*)out + g);

    // Deterministic block reduction: wave32 shuffle tree, then LDS across 8 waves
    #pragma unroll
    for (int off = 16; off > 0; off >>= 1)
        s += __shfl_xor(s, off, 32);

    __shared__ float lds[BLOCK1 / 32];
    if ((threadIdx.x & 31u) == 0u)
        lds[threadIdx.x >> 5] = s;
    __syncthreads();

    if (threadIdx.x < 32u) {
        float w = (threadIdx.x < (BLOCK1 / 32)) ? lds[threadIdx.x] : 0.0f;
        #pragma unroll
        for (int off = (BLOCK1 / 64); off > 0; off >>= 1)
            w += __shfl_xor(w, off, 32);
        if (threadIdx.x == 0u)
            partial[blockIdx.x] = w;
    }
}

// Kernel 2: single wave32 reduces the 55,296 partials (no divergence before the
// WMMA: 55,296 % 32 == 0, so EXEC is all-ones as the WMMA restriction demands).
// Final 32->1 reduction uses V_WMMA_F32_16X16X4_F32 with A = lane sums
// (VGPR0 = s, VGPR1 = 0  =>  A[i][0] = s_i (lanes 0-15), A[i][2] = s_{i+16})
// and B = all-ones 4x16 (layout-independent). D[m][n] = s_m + s_{m+16};
// summing the 8 D VGPRs per lane covers rows {r, r+8}, and lane0 + lane16
// covers all 16 rows => exact total.
__global__ __launch_bounds__(32)
void reduce_kernel(const float* __restrict__ partial, float* __restrict__ out) {
    const unsigned lane = threadIdx.x;
    float s = 0.0f;
    for (unsigned i = lane; i < (unsigned)GRID1; i += 32u)
        s += partial[i];

    v2f a; a[0] = s;    a[1] = 0.0f;
    v2f b; b[0] = 1.0f; b[1] = 1.0f;
    v8f cacc = {};
    v8f d = __builtin_amdgcn_wmma_f32_16x16x4_f32(
        /*neg_a=*/false, a, /*neg_b=*/false, b,
        /*c_mod=*/(short)0, cacc, /*reuse_a=*/false, /*reuse_b=*/false);

    float h = d[0] + d[1] + d[2] + d[3] + d[4] + d[5] + d[6] + d[7];
    float total = __shfl(h, 0, 32) + __shfl(h, 16, 32);

    if (lane == 0u)
        out[OUT_ELEMS] = sqrtf(total) / 128.0f;   // pert_norm = sqrt(sum)/N
}

extern "C" void kernel_launch(void* const* d_in, const int* in_sizes, int n_in,
                              void* d_out, int out_size, void* d_ws, size_t ws_size,
                              hipStream_t stream) {
    const int*   tok = (const int*)d_in[0];     // sentence_batch (integer -> const int*)
    const float* emb = (const float*)d_in[1];   // emb_table f32
    float* out     = (float*)d_out;             // 56,623,104 image floats + 1 scalar
    float* partial = (float*)d_ws;              // GRID1 floats = 216 KB scratch

    emb_img_kernel<<<GRID1, BLOCK1, 0, stream>>>(tok, emb, out, partial);
    reduce_kernel<<<1, 32, 0, stream>>>(partial, out);
}